// ConstantBaseLine_29592324669772
// MI455X (gfx1250) — compile-verified
//
#include <hip/hip_runtime.h>
#include <stdint.h>

#ifndef __has_builtin
#define __has_builtin(x) 0
#endif

#if __has_builtin(__builtin_amdgcn_global_load_async_to_lds_b128)
#define USE_ASYNC_LDS 1
#else
#define USE_ASYNC_LDS 0
#endif

namespace {
constexpr int kB            = 256;
constexpr int kS            = 131072;
constexpr int kTile         = 4096;
constexpr int kTilesPerRow  = kS / kTile;    // 32 -> one lane per tile in pass 2
constexpr int kThreads      = 256;           // 8 waves (wave32)
constexpr int kEPT          = kTile / kThreads;  // 16 elements per thread
constexpr int kMaskWordsPerRow = kS / 32;    // 4096 u32 words per row
}

// Builtin signature (probe-confirmed): (int4 as1*, int4 as3*, imm offset, imm cpol)
typedef int v4i __attribute__((ext_vector_type(4)));
typedef __attribute__((address_space(1))) v4i* as1_v4i_p;
typedef __attribute__((address_space(3))) v4i* as3_v4i_p;

__device__ __forceinline__ int bperm_i(int srcLane, int v) {
  return __builtin_amdgcn_ds_bpermute(srcLane << 2, v);
}
__device__ __forceinline__ float bperm_f(int srcLane, float v) {
  return __int_as_float(__builtin_amdgcn_ds_bpermute(srcLane << 2, __float_as_int(v)));
}
__device__ __forceinline__ unsigned ballot32(bool p) {
#if __has_builtin(__builtin_amdgcn_ballot_w32)
  return __builtin_amdgcn_ballot_w32(p);
#else
  return (unsigned)__ballot((int)p);
#endif
}

// ---------------------------------------------------------------------------
// Pass 1: per (row, tile): pack reset bits (1b/elem) via wave32 ballot, find
// last reset index in tile, fetch that single atten value -> (aggVal, aggHas).
// ---------------------------------------------------------------------------
__global__ void __launch_bounds__(kThreads)
pass1_agg(const float* __restrict__ att, const int* __restrict__ wet,
          unsigned* __restrict__ maskW, float* __restrict__ aggVal,
          int* __restrict__ aggHas, int useMask) {
  const int tile = blockIdx.x;
  const int row  = blockIdx.y;
  const int t    = threadIdx.x;
  const int lane = t & 31;
  const int wv   = t >> 5;
  const size_t rowOff = (size_t)row * kS;
  const int tileStart = tile * kTile;
  const int* wrow = wet + rowOff;

  int maxIdx = -1;
#pragma unroll
  for (int r = 0; r < kEPT; ++r) {
    const int s = tileStart + r * kThreads + t;      // coalesced, element order
    // WGP-scope prefetch 4 rounds (4 KB) ahead, every 4th round only
    if ((r & 3) == 0 && r + 4 < kEPT)
      __builtin_prefetch(wrow + s + 4 * kThreads, 0, 3);
    const bool reset = (s == 0) | (wrow[s] == 0);
    if (reset) maxIdx = s;
    const unsigned m = ballot32(reset);              // wave32: bit=lane
    if (useMask && lane == 0) {
      maskW[(size_t)row * kMaskWordsPerRow + (size_t)(tileStart >> 5)
            + (size_t)(r * (kThreads / 32) + wv)] = m;
    }
  }

  // wave max-reduce (butterfly via ds_bpermute)
#pragma unroll
  for (int d = 1; d < 32; d <<= 1) {
    const int o = bperm_i(lane ^ d, maxIdx);
    maxIdx = max(maxIdx, o);
  }
  __shared__ int wmax[kThreads / 32];
  if (lane == 0) wmax[wv] = maxIdx;
  __syncthreads();
  if (t == 0) {
    int mx = wmax[0];
#pragma unroll
    for (int j = 1; j < kThreads / 32; ++j) mx = max(mx, wmax[j]);
    const int gi = row * kTilesPerRow + tile;
    aggHas[gi] = (mx >= 0) ? 1 : 0;
    aggVal[gi] = (mx >= 0) ? att[rowOff + (size_t)mx] : 0.0f;  // ONE atten read/tile
  }
}

// ---------------------------------------------------------------------------
// Pass 2: one wave32 per row; 32 tile aggregates == 32 lanes. Inclusive hold
// scan, shifted to exclusive -> incoming carry value per tile.
// ---------------------------------------------------------------------------
__global__ void __launch_bounds__(32)
pass2_scan(const float* __restrict__ aggVal, const int* __restrict__ aggHas,
           float* __restrict__ carry) {
  const int row  = blockIdx.x;
  const int lane = threadIdx.x;
  const int gi   = row * kTilesPerRow + lane;
  float val = aggVal[gi];
  int   has = aggHas[gi];
#pragma unroll
  for (int d = 1; d < 32; d <<= 1) {   // combine(prev,cur)=cur.has?cur:prev
    const float pv = bperm_f(lane - d, val);
    const int   ph = bperm_i(lane - d, has);
    if (lane >= d && !has) { val = pv; has = ph; }
  }
  float ex = bperm_f(lane - 1, val);   // exclusive: carry into tile `lane`
  if (lane == 0) ex = 0.0f;            // tile 0 never consumes carry (s==0 reset)
  carry[gi] = ex;
}

// ---------------------------------------------------------------------------
// Pass 3: async-stream atten tile to LDS; overlap mask/carry global loads with
// the transfer; blocked hold-scan; float4 output stores.
// ---------------------------------------------------------------------------
__global__ void __launch_bounds__(kThreads)
pass3_apply(const float* __restrict__ att, const int* __restrict__ wet,
            const unsigned* __restrict__ maskW, const float* __restrict__ carry,
            float* __restrict__ out, int useMask) {
  const int tile = blockIdx.x;
  const int row  = blockIdx.y;
  const int t    = threadIdx.x;
  const int lane = t & 31;
  const int wv   = t >> 5;
  const size_t rowOff = (size_t)row * kS;
  const int tileStart = tile * kTile;
  const float* arow = att + rowOff;
  const int base = tileStart + t * kEPT;   // this thread's 16 contiguous elems

#if USE_ASYNC_LDS
  __shared__ float tileBuf[kTile];         // 16 KB staging
  const float* gsrc = arow + base;         // 64B aligned
  float*       ldst = &tileBuf[t * kEPT];
#pragma unroll
  for (int k = 0; k < 4; ++k) {            // 4 x b128 per lane -> LDS
    __builtin_amdgcn_global_load_async_to_lds_b128(
        (as1_v4i_p)(gsrc + 4 * k), (as3_v4i_p)(ldst + 4 * k), 0, 0);
  }
#endif

  // Independent global loads issued while the async engine streams the tile:
  // 16 reset bits for this thread's elements, and this tile's incoming carry.
  unsigned mask16;
  if (useMask) {
    const unsigned w = maskW[(size_t)row * kMaskWordsPerRow + (size_t)(base >> 5)];
    mask16 = (w >> ((t & 1) * 16)) & 0xFFFFu;
  } else {
    const int* wrow = wet + rowOff;
    mask16 = 0;
#pragma unroll
    for (int e = 0; e < kEPT; ++e) {
      const int s = base + e;
      const bool reset = (s == 0) | (wrow[s] == 0);
      mask16 |= (unsigned)reset << e;
    }
  }
  const float carryVal = carry[row * kTilesPerRow + tile];

  float v[kEPT];
#if USE_ASYNC_LDS
  asm volatile("s_wait_asynccnt 0" ::: "memory");
  {
    const float4* lp = (const float4*)ldst;
#pragma unroll
    for (int k = 0; k < 4; ++k) {
      const float4 q = lp[k];
      v[4 * k + 0] = q.x; v[4 * k + 1] = q.y; v[4 * k + 2] = q.z; v[4 * k + 3] = q.w;
    }
  }
#else
  {
    const float4* gp = (const float4*)(arow + base);
#pragma unroll
    for (int k = 0; k < 4; ++k) {
      const float4 q = gp[k];
      v[4 * k + 0] = q.x; v[4 * k + 1] = q.y; v[4 * k + 2] = q.z; v[4 * k + 3] = q.w;
    }
  }
#endif

  // thread-local aggregate: value at highest set bit
  float lval = 0.0f;
  const int lhas = (mask16 != 0u);
#pragma unroll
  for (int e = 0; e < kEPT; ++e)
    if (mask16 & (1u << e)) lval = v[e];

  // wave32 inclusive hold-scan of thread aggregates
  float sval = lval; int shas = lhas;
#pragma unroll
  for (int d = 1; d < 32; d <<= 1) {
    const float pv = bperm_f(lane - d, sval);
    const int   ph = bperm_i(lane - d, shas);
    if (lane >= d && !shas) { sval = pv; shas = ph; }
  }
  float exVal = bperm_f(lane - 1, sval);
  int   exHas = bperm_i(lane - 1, shas);
  if (lane == 0) { exVal = 0.0f; exHas = 0; }

  __shared__ float wAggV[kThreads / 32];
  __shared__ int   wAggH[kThreads / 32];
  if (lane == 31) { wAggV[wv] = sval; wAggH[wv] = shas; }
  __syncthreads();

  float pfx = carryVal;                          // last reset before this tile
  for (int j = wv - 1; j >= 0; --j) {
    if (wAggH[j]) { pfx = wAggV[j]; break; }     // nearest earlier wave with a reset
  }
  if (exHas) pfx = exVal;                        // nearest earlier thread in-wave wins

  // serial hold over this thread's 16 elements; float4 stores
  float* orow = out + rowOff;
  float run = pfx;
#pragma unroll
  for (int k = 0; k < 4; ++k) {
    float4 o;
    if (mask16 & (1u << (4 * k + 0))) run = v[4 * k + 0]; o.x = run;
    if (mask16 & (1u << (4 * k + 1))) run = v[4 * k + 1]; o.y = run;
    if (mask16 & (1u << (4 * k + 2))) run = v[4 * k + 2]; o.z = run;
    if (mask16 & (1u << (4 * k + 3))) run = v[4 * k + 3]; o.w = run;
    ((float4*)(orow + base))[k] = o;
  }
}

// ---------------------------------------------------------------------------
extern "C" void kernel_launch(void* const* d_in, const int* in_sizes, int n_in,
                              void* d_out, int out_size, void* d_ws, size_t ws_size,
                              hipStream_t stream) {
  const float* att = (const float*)d_in[0];   // [B,S] float32
  const int*   wet = (const int*)d_in[1];     // [B,S] bool as int
  float*       out = (float*)d_out;           // [B,S] float32

  const size_t maskWords = (size_t)kB * kMaskWordsPerRow;       // 1M words = 4 MB
  const size_t nAgg      = (size_t)kB * kTilesPerRow;           // 8192
  const size_t needMask  = maskWords * sizeof(unsigned) + nAgg * 12;
  const int useMask = (ws_size >= needMask) ? 1 : 0;

  char* ws = (char*)d_ws;
  unsigned* maskW = nullptr;
  if (useMask) { maskW = (unsigned*)ws; ws += maskWords * sizeof(unsigned); }
  float* aggVal = (float*)ws; ws += nAgg * sizeof(float);
  int*   aggHas = (int*)ws;   ws += nAgg * sizeof(int);
  float* carry  = (float*)ws;

  const dim3 grid(kTilesPerRow, kB);
  pass1_agg<<<grid, kThreads, 0, stream>>>(att, wet, maskW, aggVal, aggHas, useMask);
  pass2_scan<<<kB, 32, 0, stream>>>(aggVal, aggHas, carry);
  pass3_apply<<<grid, kThreads, 0, stream>>>(att, wet, maskW, carry, out, useMask);
}